// traj_modif_17343077941290
// MI455X (gfx1250) — compile-verified
//
#include <hip/hip_runtime.h>
#include <math.h>

typedef float v2f __attribute__((ext_vector_type(2)));
typedef float v8f __attribute__((ext_vector_type(8)));

#define LMBDA 0.1f

// ---------------------------------------------------------------------------
// K0: traj_n = traj + delta -> d_out; build packed WMMA A-operand table;
//     init global min slot.
// packedA[r*32 + lane] is exactly the v2f the WMMA A operand needs:
//   lane<16  (K=0,1): (-2*tx, -2*ty) of row r*16+lane
//   lane>=16 (K=2,3): (tx^2+ty^2, 1.0)
// ---------------------------------------------------------------------------
__global__ void prep_kernel(const float* __restrict__ traj,
                            const float* __restrict__ delta,
                            float* __restrict__ out_trajn,
                            v2f* __restrict__ packedA,
                            unsigned int* __restrict__ minslot,
                            int N, int numRowTiles) {
  const int t = blockIdx.x * blockDim.x + threadIdx.x;
  const int stride = blockDim.x * gridDim.x;
  if (t == 0) *minslot = 0x7F7FFFFFu;  // FLT_MAX bit pattern

  const int total = 2 * N;
  for (int i = t; i < total; i += stride)
    out_trajn[i] = traj[i] + delta[i];

  const int totalA = numRowTiles * 32;
  for (int i = t; i < totalA; i += stride) {
    const int lane = i & 31;
    const int row  = (i >> 5) * 16 + (lane & 15);
    v2f a;
    if (row < N) {
      const float tx = traj[2 * row]     + delta[2 * row];
      const float ty = traj[2 * row + 1] + delta[2 * row + 1];
      if (lane < 16) { a.x = -2.0f * tx;       a.y = -2.0f * ty; }
      else           { a.x = tx * tx + ty * ty; a.y = 1.0f; }
    } else {  // OOB row -> enormous dist^2, never the min
      if (lane < 16) { a.x = 0.0f;  a.y = 0.0f; }
      else           { a.x = 3e38f; a.y = 1.0f; }
    }
    packedA[i] = a;
  }
}

// ---------------------------------------------------------------------------
// K1: global min of pairwise dist^2 via V_WMMA_F32_16X16X4_F32.
// dist^2 = ||t||^2 - 2 t.b + ||b||^2 packed into K=4 (see prep).
// Each wave holds TWO border column tiles in registers (b0,b1) and sweeps all
// row tiles; one coalesced b64 A-load feeds two WMMAs per iteration.
// ---------------------------------------------------------------------------
__global__ __launch_bounds__(256)
void min_dist_wmma_kernel(const v2f* __restrict__ packedA,   // [numRowTiles*32]
                          const float* __restrict__ border,  // [M,2]
                          unsigned int* __restrict__ minslot,
                          int numRowTiles, int M) {
  const int wave = threadIdx.x >> 5;
  const int lane = threadIdx.x & 31;
  const int pairIdx = blockIdx.x * (blockDim.x >> 5) + wave;

  const int tile0 = pairIdx * 2;
  const int tile1 = pairIdx * 2 + 1;

  // --- B operands: two border column tiles, resident for the whole sweep ---
  auto makeB = [&](int tile) -> v2f {
    const int col = tile * 16 + (lane & 15);
    float bx = 1e18f, by = 0.0f;  // OOB col -> enormous dist^2
    if (col < M) { bx = border[2 * col]; by = border[2 * col + 1]; }
    v2f b;
    if (lane < 16) { b.x = bx;   b.y = by; }                 // K = 0,1
    else           { b.x = 1.0f; b.y = bx * bx + by * by; }  // K = 2,3
    return b;
  };
  const v2f b0 = makeB(tile0);
  const v2f b1 = makeB(tile1);

  float rmin = 3.0e38f;
  const v2f* aptr = packedA + lane;

#pragma unroll 4
  for (int r = 0; r < numRowTiles; ++r) {
    const v2f a = aptr[r * 32];  // coalesced 256B per wave-iteration

    v8f c0 = {};
    v8f d0 = __builtin_amdgcn_wmma_f32_16x16x4_f32(
        false, a, false, b0, (short)0, c0, false, false);
    v8f c1 = {};
    v8f d1 = __builtin_amdgcn_wmma_f32_16x16x4_f32(
        false, a, false, b1, (short)0, c1, false, false);

    const float m0 = fminf(fminf(fminf(d0[0], d0[1]), fminf(d0[2], d0[3])),
                           fminf(fminf(d0[4], d0[5]), fminf(d0[6], d0[7])));
    const float m1 = fminf(fminf(fminf(d1[0], d1[1]), fminf(d1[2], d1[3])),
                           fminf(fminf(d1[4], d1[5]), fminf(d1[6], d1[7])));
    rmin = fminf(rmin, fminf(m0, m1));
  }

  // wave32 min-reduce
  for (int off = 16; off >= 1; off >>= 1)
    rmin = fminf(rmin, __shfl_xor(rmin, off, 32));

  if (lane == 0) {
    rmin = fmaxf(rmin, 0.0f);  // dist^2 >= 0; keep uint ordering valid
    atomicMin(minslot, __float_as_uint(rmin));
  }
}

// ---------------------------------------------------------------------------
// K2: l2 = ||delta||_F, loss = l2 + LMBDA * exp(-sqrt(min dist^2))
// ---------------------------------------------------------------------------
__global__ __launch_bounds__(256)
void finalize_kernel(const float* __restrict__ delta,
                     const unsigned int* __restrict__ minslot,
                     float* __restrict__ out_loss,
                     int total /* N*2 */) {
  __shared__ float sdata[256];
  float s = 0.0f;
  for (int i = threadIdx.x; i < total; i += blockDim.x) {
    float v = delta[i];
    s += v * v;
  }
  sdata[threadIdx.x] = s;
  __syncthreads();
  for (int stride = 128; stride > 0; stride >>= 1) {
    if ((int)threadIdx.x < stride) sdata[threadIdx.x] += sdata[threadIdx.x + stride];
    __syncthreads();
  }
  if (threadIdx.x == 0) {
    float l2   = sqrtf(sdata[0]);
    float dmin = sqrtf(__uint_as_float(*minslot));
    *out_loss = l2 + LMBDA * expf(-dmin);
  }
}

// ---------------------------------------------------------------------------
extern "C" void kernel_launch(void* const* d_in, const int* in_sizes, int n_in,
                              void* d_out, int out_size, void* d_ws, size_t ws_size,
                              hipStream_t stream) {
  const float* traj   = (const float*)d_in[0];
  const float* border = (const float*)d_in[1];
  const float* delta  = (const float*)d_in[2];
  float* out = (float*)d_out;

  const int N = in_sizes[0] / 2;   // 1024
  const int M = in_sizes[1] / 2;   // 262144

  const int numRowTiles = (N + 15) / 16;
  const int numColTiles = (M + 15) / 16;

  unsigned int* minslot = (unsigned int*)d_ws;
  v2f* packedA = (v2f*)((char*)d_ws + 16);   // numRowTiles*32 v2f = 16KB @ N=1024

  prep_kernel<<<16, 256, 0, stream>>>(traj, delta, out, packedA, minslot,
                                      N, numRowTiles);

  const int numPairs      = (numColTiles + 1) / 2;
  const int wavesPerBlock = 256 / 32;
  const int blocks        = (numPairs + wavesPerBlock - 1) / wavesPerBlock;
  min_dist_wmma_kernel<<<blocks, 256, 0, stream>>>(packedA, border, minslot,
                                                   numRowTiles, M);

  finalize_kernel<<<1, 256, 0, stream>>>(delta, minslot, out + N * 2, N * 2);
}